// PointConvDensity_83339545411690
// MI455X (gfx1250) — compile-verified
//
#include <hip/hip_runtime.h>
#include <hip/hip_bf16.h>

// ---------------------------------------------------------------------------
// PointConvDensity for MI455X (gfx1250, wave32, WMMA).
// Shapes fixed by setup_inputs(): B=4, N=4096, K=32, C=64, O=16.
// Bandwidth-bound (~211MB essential traffic -> ~9us @ 23.3TB/s), so all
// matrix math uses exact FP32 WMMA (V_WMMA_F32_16X16X4_F32).
// Raw v_exp_f32 in the density loop; async global->LDS staging
// (GLOBAL_LOAD_ASYNC_TO_LDS_B128, ASYNCcnt) overlapped with the w-MLP in
// the final einsum kernel.
// ---------------------------------------------------------------------------

typedef __attribute__((ext_vector_type(2))) float v2f;
typedef __attribute__((ext_vector_type(8))) float v8f;
typedef int v4i_vs __attribute__((vector_size(16)));   // matches builtin param

#define BB   4
#define NN   4096
#define KK   32
#define CC   64
#define OO   16
#define NKP  (BB * NN)            // 16384 positions for d-branch BN
#define PKP  (BB * NN * KK)       // 524288 positions for w-branch BN
#define EPSV 1e-5f

#if defined(__has_builtin)
#  if __has_builtin(__builtin_amdgcn_global_load_async_to_lds_b128) && \
      __has_builtin(__builtin_amdgcn_s_wait_asynccnt)
#    define HAVE_ASYNC_LDS 1
#  endif
#endif
#ifndef HAVE_ASYNC_LDS
#  define HAVE_ASYNC_LDS 0
#endif

__device__ __forceinline__ float fast_exp2(float x)
{
#if defined(__has_builtin)
#  if __has_builtin(__builtin_amdgcn_exp2f)
    return __builtin_amdgcn_exp2f(x);   // raw v_exp_f32, no denorm fixups
#  else
    return exp2f(x);
#  endif
#else
    return exp2f(x);
#endif
}

struct DnParams {
    const float *w0, *b0, *g0, *be0;
    const float *w1, *b1, *g1, *be1;
    const float *w2, *b2, *g2, *be2;
};
struct WnParams {
    const float *w0, *b0, *g0, *be0;
    const float *w1, *b1, *g1, *be1;
    const float *w2, *b2, *g2, *be2;
};

// ---------------------------------------------------------------------------
// Kernel 1: density.  dens[b][n] = (1/1024) * sum_m exp2( -72.1347*dist(n,m) )
// Gram tiles via V_WMMA_F32_16X16X4_F32 with K padded 3->4.
// A layout (16x4 f32): lanes 0-15 hold (K=0,K=1)=(x,y); lanes 16-31 (K=2,K=3)=(z,0).
// B layout (4x16 f32): mirrored.  C: VGPR v -> row v+8*half, lane&15 -> col.
// ---------------------------------------------------------------------------
__global__ __launch_bounds__(128) void density_kernel(
    const float* __restrict__ xyz, float* __restrict__ dens)
{
    __shared__ float sx[NN], sy[NN], sz[NN];   // 48 KB of 320 KB/WGP
    const int b      = blockIdx.x >> 6;        // 64 blocks per batch
    const int n_base = (blockIdx.x & 63) * 64; // 64 rows per block (4 waves x 16)
    const float* xb  = xyz + (size_t)b * NN * 3;

    for (int i = threadIdx.x; i < NN; i += 128) {
        sx[i] = xb[i * 3 + 0];
        sy[i] = xb[i * 3 + 1];
        sz[i] = xb[i * 3 + 2];
    }
    __syncthreads();

    const int lane = threadIdx.x & 31;
    const int wave = threadIdx.x >> 5;
    const int half = lane >> 4;
    const int l15  = lane & 15;
    const int n0   = n_base + wave * 16;

    const int ar = n0 + l15;
    v2f a;
    a.x = half ? sz[ar] : sx[ar];
    a.y = half ? 0.0f   : sy[ar];

    float sqn[8];
#pragma unroll
    for (int v = 0; v < 8; ++v) {
        const int r = n0 + v + 8 * half;
        sqn[v] = sx[r] * sx[r] + sy[r] * sy[r] + sz[r] * sz[r];
    }

    v8f acc = {};
    const float KE = -72.13475204444817f;  // -(1/(2*0.1^2)) * log2(e)

    for (int m0 = 0; m0 < NN; m0 += 16) {
        const int mc = m0 + l15;
        const float xm = sx[mc], ym = sy[mc], zm = sz[mc];
        v2f bf;
        bf.x = half ? zm : xm;
        bf.y = half ? 0.0f : ym;
        v8f gzero = {};
        v8f gram = __builtin_amdgcn_wmma_f32_16x16x4_f32(
            false, a, false, bf, (short)0, gzero, false, false);
        const float sqm = xm * xm + ym * ym + zm * zm;
#pragma unroll
        for (int v = 0; v < 8; ++v) {
            const float dist = sqn[v] + sqm - 2.0f * gram[v];
            acc[v] += fast_exp2(dist * KE);
        }
    }

#pragma unroll
    for (int v = 0; v < 8; ++v) {
        float s = acc[v];
        s += __shfl_xor(s, 1, 32);
        s += __shfl_xor(s, 2, 32);
        s += __shfl_xor(s, 4, 32);
        s += __shfl_xor(s, 8, 32);   // stays within each 16-lane half
        if (l15 == 0)
            dens[(size_t)b * NN + n0 + v + 8 * half] = s * (1.0f / 1024.0f);
    }
}

// ---------------------------------------------------------------------------
// Kernel 2: d-branch MLP (1->8->8->1, BN over B*N, relu), single block,
// deterministic LDS tree reductions, cheap chain recompute per phase.
// ---------------------------------------------------------------------------
__global__ __launch_bounds__(512) void dn_mlp_kernel(
    const float* __restrict__ dens, float* __restrict__ ds, DnParams P)
{
    __shared__ float red[512][16];  // 32 KB
    __shared__ float statb[16];
    __shared__ float sW1[64];

    const int tid = threadIdx.x;
    if (tid < 64) sW1[tid] = P.w1[tid];
    __syncthreads();

    const float inv = 1.0f / (float)NKP;

    auto blkreduce = [&](float* part, int nv) {
        for (int v = 0; v < nv; ++v) red[tid][v] = part[v];
        __syncthreads();
        for (int s = 256; s > 0; s >>= 1) {
            if (tid < s)
                for (int v = 0; v < nv; ++v) red[tid][v] += red[tid + s][v];
            __syncthreads();
        }
        if (tid == 0)
            for (int v = 0; v < nv; ++v) statb[v] = red[0][v];
        __syncthreads();
    };

    // Phase A: E[x], E[x^2] of dens; layer0 is affine in x -> stats analytic.
    float part[16];
    part[0] = 0.0f; part[1] = 0.0f;
    for (int p = tid; p < NKP; p += 512) {
        const float x = dens[p];
        part[0] += x; part[1] += x * x;
    }
    blkreduce(part, 2);
    const float mx = statb[0] * inv;
    const float vx = statb[1] * inv - mx * mx;

    float A0[8], B0r[8], S0[8], T0[8], B1r[8];
#pragma unroll
    for (int o = 0; o < 8; ++o) {
        A0[o]  = P.w0[o];
        B0r[o] = P.b0[o];
        B1r[o] = P.b1[o];
        const float m = A0[o] * mx + B0r[o];
        const float v = A0[o] * A0[o] * vx;
        const float s = P.g0[o] * rsqrtf(v + EPSV);
        S0[o] = s;
        T0[o] = P.be0[o] - m * s;
    }

    // Phase B: stats of y1
    for (int v = 0; v < 16; ++v) part[v] = 0.0f;
    for (int p = tid; p < NKP; p += 512) {
        const float x = dens[p];
        float z0[8];
#pragma unroll
        for (int o = 0; o < 8; ++o)
            z0[o] = fmaxf(0.0f, (A0[o] * x + B0r[o]) * S0[o] + T0[o]);
#pragma unroll
        for (int o = 0; o < 8; ++o) {
            float y1 = B1r[o];
#pragma unroll
            for (int i = 0; i < 8; ++i) y1 += sW1[o * 8 + i] * z0[i];
            part[o] += y1;
            part[8 + o] += y1 * y1;
        }
    }
    blkreduce(part, 16);
    float S1[8], T1[8];
#pragma unroll
    for (int o = 0; o < 8; ++o) {
        const float m = statb[o] * inv;
        const float v = statb[8 + o] * inv - m * m;
        const float s = P.g1[o] * rsqrtf(v + EPSV);
        S1[o] = s;
        T1[o] = P.be1[o] - m * s;
    }

    // Phase C: stats of y2 (scalar channel)
    float W2r[8];
#pragma unroll
    for (int i = 0; i < 8; ++i) W2r[i] = P.w2[i];
    const float b2v = P.b2[0];
    part[0] = 0.0f; part[1] = 0.0f;
    for (int p = tid; p < NKP; p += 512) {
        const float x = dens[p];
        float z0[8], z1[8];
#pragma unroll
        for (int o = 0; o < 8; ++o)
            z0[o] = fmaxf(0.0f, (A0[o] * x + B0r[o]) * S0[o] + T0[o]);
#pragma unroll
        for (int o = 0; o < 8; ++o) {
            float y1 = B1r[o];
#pragma unroll
            for (int i = 0; i < 8; ++i) y1 += sW1[o * 8 + i] * z0[i];
            z1[o] = fmaxf(0.0f, y1 * S1[o] + T1[o]);
        }
        float y2 = b2v;
#pragma unroll
        for (int i = 0; i < 8; ++i) y2 += W2r[i] * z1[i];
        part[0] += y2; part[1] += y2 * y2;
    }
    blkreduce(part, 2);
    const float m2 = statb[0] * inv;
    const float v2 = statb[1] * inv - m2 * m2;
    const float S2 = P.g2[0] * rsqrtf(v2 + EPSV);
    const float T2 = P.be2[0] - m2 * S2;

    // Phase D: write ds
    for (int p = tid; p < NKP; p += 512) {
        const float x = dens[p];
        float z0[8], z1[8];
#pragma unroll
        for (int o = 0; o < 8; ++o)
            z0[o] = fmaxf(0.0f, (A0[o] * x + B0r[o]) * S0[o] + T0[o]);
#pragma unroll
        for (int o = 0; o < 8; ++o) {
            float y1 = B1r[o];
#pragma unroll
            for (int i = 0; i < 8; ++i) y1 += sW1[o * 8 + i] * z0[i];
            z1[o] = fmaxf(0.0f, y1 * S1[o] + T1[o]);
        }
        float y2 = b2v;
#pragma unroll
        for (int i = 0; i < 8; ++i) y2 += W2r[i] * z1[i];
        ds[p] = fmaxf(0.0f, y2 * S2 + T2);
    }
}

// ---------------------------------------------------------------------------
// w-branch chain evaluator (compile-time depth).
// UPTO: 0 -> y0[8], 1 -> y1[8], 2 -> y2[16], 3 -> z2[16] (final relu'd).
// ---------------------------------------------------------------------------
template <int UPTO>
__device__ __forceinline__ void wn_eval(
    const float gx[3],
    const float* __restrict__ W0, const float* __restrict__ B0,
    const float* __restrict__ W1, const float* __restrict__ B1,
    const float* __restrict__ W2, const float* __restrict__ B2,
    const float* __restrict__ S0, const float* __restrict__ T0,
    const float* __restrict__ S1, const float* __restrict__ T1,
    const float* __restrict__ S2, const float* __restrict__ T2,
    float* __restrict__ out)
{
    float y0[8];
#pragma unroll
    for (int o = 0; o < 8; ++o)
        y0[o] = W0[o * 3 + 0] * gx[0] + W0[o * 3 + 1] * gx[1] +
                W0[o * 3 + 2] * gx[2] + B0[o];
    if constexpr (UPTO == 0) {
#pragma unroll
        for (int o = 0; o < 8; ++o) out[o] = y0[o];
    } else {
        float z0[8];
#pragma unroll
        for (int o = 0; o < 8; ++o)
            z0[o] = fmaxf(0.0f, y0[o] * S0[o] + T0[o]);
        float y1[8];
#pragma unroll
        for (int o = 0; o < 8; ++o) {
            float s = B1[o];
#pragma unroll
            for (int i = 0; i < 8; ++i) s += W1[o * 8 + i] * z0[i];
            y1[o] = s;
        }
        if constexpr (UPTO == 1) {
#pragma unroll
            for (int o = 0; o < 8; ++o) out[o] = y1[o];
        } else {
            float z1[8];
#pragma unroll
            for (int o = 0; o < 8; ++o)
                z1[o] = fmaxf(0.0f, y1[o] * S1[o] + T1[o]);
            float y2[16];
#pragma unroll
            for (int o = 0; o < 16; ++o) {
                float s = B2[o];
#pragma unroll
                for (int i = 0; i < 8; ++i) s += W2[o * 8 + i] * z1[i];
                y2[o] = s;
            }
            if constexpr (UPTO == 2) {
#pragma unroll
                for (int o = 0; o < 16; ++o) out[o] = y2[o];
            } else {
#pragma unroll
                for (int o = 0; o < 16; ++o)
                    out[o] = fmaxf(0.0f, y2[o] * S2[o] + T2[o]);
            }
        }
    }
}

// ---------------------------------------------------------------------------
// Kernel 3: w-branch BN stats pass for layer L.  Each thread accumulates 4
// positions, per-block tree reduction -> per-block partials; a 1-block
// reduce kernel sums the 512 partials deterministically.
// stats layout (raw sums): [0..15]=L0 sum/sumsq, [16..31]=L1, [32..63]=L2.
// ---------------------------------------------------------------------------
template <int L>
__global__ __launch_bounds__(256) void wn_stats_kernel(
    const float* __restrict__ gxn, const int* __restrict__ idx,
    const float* __restrict__ ds, const float* __restrict__ stats,
    float* __restrict__ partials, WnParams P)
{
    constexpr int NCH = (L == 2) ? 16 : 8;
    constexpr int NV  = 2 * NCH;

    __shared__ float red[256][32];  // 32 KB
    __shared__ float sW0[24], sB0[8], sW1[64], sB1[8], sW2[128], sB2[16];
    __shared__ float sS0[8], sT0[8], sS1[8], sT1[8];

    const int tid = threadIdx.x;
    if (tid < 24) sW0[tid] = P.w0[tid];
    if (tid < 8)  { sB0[tid] = P.b0[tid]; sB1[tid] = P.b1[tid]; }
    if (tid < 64) sW1[tid] = P.w1[tid];
    if (tid < 128) sW2[tid] = P.w2[tid];
    if (tid < 16) sB2[tid] = P.b2[tid];

    const float invc = 1.0f / (float)PKP;
    if (L >= 1 && tid < 8) {
        const float m = stats[tid] * invc;
        const float v = stats[8 + tid] * invc - m * m;
        const float s = P.g0[tid] * rsqrtf(v + EPSV);
        sS0[tid] = s;
        sT0[tid] = P.be0[tid] - m * s;
    }
    if (L >= 2 && tid < 8) {
        const float m = stats[16 + tid] * invc;
        const float v = stats[24 + tid] * invc - m * m;
        const float s = P.g1[tid] * rsqrtf(v + EPSV);
        sS1[tid] = s;
        sT1[tid] = P.be1[tid] - m * s;
    }
    __syncthreads();

    float part[32];
#pragma unroll
    for (int v = 0; v < NV; ++v) part[v] = 0.0f;

#pragma unroll
    for (int j = 0; j < 4; ++j) {
        const int g   = blockIdx.x * 1024 + j * 256 + tid;  // (b,n,k) enum
        const int b   = g >> 17;                            // N*K = 2^17
        const int rem = g & (NN * KK - 1);
        const int n   = rem >> 5;
        const int k   = rem & 31;

        const float gd = ds[(size_t)b * NN + idx[g]];
        float gx[3];
#pragma unroll
        for (int i = 0; i < 3; ++i)
            gx[i] = gxn[(size_t)(b * 3 + i) * (NN * KK) + (size_t)n * KK + k] * gd;

        float y[16];
        wn_eval<L>(gx, sW0, sB0, sW1, sB1, sW2, sB2,
                   sS0, sT0, sS1, sT1, nullptr, nullptr, y);
#pragma unroll
        for (int o = 0; o < NCH; ++o) {
            part[o]       += y[o];
            part[NCH + o] += y[o] * y[o];
        }
    }

    for (int v = 0; v < NV; ++v) red[tid][v] = part[v];
    __syncthreads();
    for (int s = 128; s > 0; s >>= 1) {
        if (tid < s)
            for (int v = 0; v < NV; ++v) red[tid][v] += red[tid + s][v];
        __syncthreads();
    }
    if (tid < NV) partials[(size_t)blockIdx.x * 32 + tid] = red[0][tid];
}

__global__ __launch_bounds__(64) void wn_reduce_kernel(
    const float* __restrict__ partials, float* __restrict__ stats,
    int off, int nv, int nblocks)
{
    const int v = threadIdx.x;
    if (v < nv) {
        float s = 0.0f;
        for (int i = 0; i < nblocks; ++i) s += partials[(size_t)i * 32 + v];
        stats[off + v] = s;
    }
}

// ---------------------------------------------------------------------------
// Kernel 4: final fused w-eval + einsum.  One block per (b,n).
// GEMM: out[64x16] = gp[64x32] * w^T[32x16] via 8 chained f32 WMMA K=4.
// gp tile staged via GLOBAL_LOAD_ASYNC_TO_LDS_B128 (ASYNCcnt), overlapped
// with the per-point w-MLP; LDS rows padded to stride 36 floats
// (16B-aligned, conflict-free).
// ---------------------------------------------------------------------------
__global__ __launch_bounds__(128) void final_einsum_kernel(
    const float* __restrict__ gp, const float* __restrict__ gxn,
    const int* __restrict__ idx, const float* __restrict__ ds,
    const float* __restrict__ stats, float* __restrict__ out, WnParams P)
{
    __shared__ float sgp[64 * 36];  // 9216 B
    __shared__ float sw[16 * 36];   // 2304 B
    __shared__ float sW0[24], sB0[8], sW1[64], sB1[8], sW2[128], sB2[16];
    __shared__ float sS0[8], sT0[8], sS1[8], sT1[8], sS2[16], sT2[16];

    const int tid = threadIdx.x;
    const int b   = blockIdx.x >> 12;
    const int n   = blockIdx.x & (NN - 1);

#if HAVE_ASYNC_LDS
    // Kick off the 8KB gp tile DMA first; overlap with param setup + w-MLP.
    for (int u = tid; u < 512; u += 128) {
        const int row = u >> 3, q = u & 7;
        const float* gsrc = gp + (size_t)(b * CC + row) * (NN * KK) +
                            (size_t)n * KK + q * 4;
        __builtin_amdgcn_global_load_async_to_lds_b128(
            (__attribute__((address_space(1))) v4i_vs*)gsrc,
            (__attribute__((address_space(3))) v4i_vs*)(&sgp[row * 36 + q * 4]),
            0, 0);
    }
#endif

    if (tid < 24) sW0[tid] = P.w0[tid];
    if (tid < 8)  { sB0[tid] = P.b0[tid]; sB1[tid] = P.b1[tid]; }
    if (tid < 64) sW1[tid] = P.w1[tid];
    if (tid < 128) sW2[tid] = P.w2[tid];
    if (tid < 16) sB2[tid] = P.b2[tid];

    const float invc = 1.0f / (float)PKP;
    if (tid < 8) {
        float m = stats[tid] * invc;
        float v = stats[8 + tid] * invc - m * m;
        float s = P.g0[tid] * rsqrtf(v + EPSV);
        sS0[tid] = s; sT0[tid] = P.be0[tid] - m * s;
        m = stats[16 + tid] * invc;
        v = stats[24 + tid] * invc - m * m;
        s = P.g1[tid] * rsqrtf(v + EPSV);
        sS1[tid] = s; sT1[tid] = P.be1[tid] - m * s;
    }
    if (tid < 16) {
        const float m = stats[32 + tid] * invc;
        const float v = stats[48 + tid] * invc - m * m;
        const float s = P.g2[tid] * rsqrtf(v + EPSV);
        sS2[tid] = s; sT2[tid] = P.be2[tid] - m * s;
    }

#if !HAVE_ASYNC_LDS
    // Fallback: synchronous staging, coalesced float4 (row = cacheline).
    for (int u = tid; u < 512; u += 128) {
        const int row = u >> 3, q = u & 7;
        const float4 val = *(const float4*)(gp +
            (size_t)(b * CC + row) * (NN * KK) + (size_t)n * KK + q * 4);
        *(float4*)(&sgp[row * 36 + q * 4]) = val;
    }
#endif
    __syncthreads();   // params (and fallback sgp) visible

    if (tid < 32) {
        const int k = tid;
        const float gd =
            ds[(size_t)b * NN + idx[((size_t)b * NN + n) * KK + k]];
        float gx[3];
#pragma unroll
        for (int i = 0; i < 3; ++i)
            gx[i] = gxn[(size_t)(b * 3 + i) * (NN * KK) + (size_t)n * KK + k] * gd;
        float z2[16];
        wn_eval<3>(gx, sW0, sB0, sW1, sB1, sW2, sB2,
                   sS0, sT0, sS1, sT1, sS2, sT2, z2);
#pragma unroll
        for (int o = 0; o < 16; ++o) sw[o * 36 + k] = z2[o];
    }

#if HAVE_ASYNC_LDS
    __builtin_amdgcn_s_wait_asynccnt(0);  // each wave's DMA landed in LDS
#endif
    __syncthreads();   // sw + all waves' sgp visible

    const int lane = tid & 31;
    const int wave = tid >> 5;
    const int half = lane >> 4;
    const int l15  = lane & 15;
    const int c0   = wave * 16;
    const int cr   = c0 + l15;

    v8f acc = {};
#pragma unroll
    for (int j = 0; j < 8; ++j) {
        const int k0 = j * 4;
        // A: lanes 0-15 -> gp[c][k0..k0+1], lanes 16-31 -> gp[c][k0+2..k0+3]
        const v2f a = *(const v2f*)(&sgp[cr * 36 + k0 + 2 * half]);
        // B[k][o] = w[o][k]: lane l<16 -> (w[l][k0],w[l][k0+1]); l+16 -> (+2,+3)
        const v2f bf = *(const v2f*)(&sw[l15 * 36 + k0 + 2 * half]);
        acc = __builtin_amdgcn_wmma_f32_16x16x4_f32(
            false, a, false, bf, (short)0, acc, false, false);
    }

#pragma unroll
    for (int v = 0; v < 8; ++v) {
        const int c = c0 + v + 8 * half;
        out[((size_t)(b * CC + c) * NN + n) * OO + l15] = acc[v];
    }
}

// ---------------------------------------------------------------------------
// Host launcher
// ---------------------------------------------------------------------------
extern "C" void kernel_launch(void* const* d_in, const int* in_sizes, int n_in,
                              void* d_out, int out_size, void* d_ws, size_t ws_size,
                              hipStream_t stream)
{
    const float* xyz = (const float*)d_in[0];
    const float* gxn = (const float*)d_in[1];
    const int*   idx = (const int*)d_in[2];
    const float* gp  = (const float*)d_in[3];

    // Canonical (insertion) param order:
    // dn_w0,dn_b0,dn_g0,dn_be0, dn_w1..., dn_w2..., wn_w0..., wn_w1..., wn_w2...
    const float* prm[24];
    if (n_in >= 28) {
        const bool alpha = (in_sizes[6] == 1);  // alphabetical flatten: idx6 = dn_b2 (size 1)
        if (!alpha) {
            for (int j = 0; j < 24; ++j) prm[j] = (const float*)d_in[4 + j];
        } else {
            static const int amap[24] = { 9, 0, 6, 3, 10, 1, 7, 4, 11, 2, 8, 5,
                                          21, 12, 18, 15, 22, 13, 19, 16, 23, 14, 20, 17 };
            for (int j = 0; j < 24; ++j) prm[j] = (const float*)d_in[4 + amap[j]];
        }
    } else {
        // Single concatenated params buffer, insertion order.
        static const int sz[24] = { 8, 8, 8, 8, 64, 8, 8, 8, 8, 1, 1, 1,
                                    24, 8, 8, 8, 64, 8, 8, 8, 128, 16, 16, 16 };
        const float* base = (const float*)d_in[4];
        size_t off = 0;
        for (int j = 0; j < 24; ++j) { prm[j] = base + off; off += sz[j]; }
    }

    DnParams dp = { prm[0], prm[1], prm[2], prm[3],
                    prm[4], prm[5], prm[6], prm[7],
                    prm[8], prm[9], prm[10], prm[11] };
    WnParams wp = { prm[12], prm[13], prm[14], prm[15],
                    prm[16], prm[17], prm[18], prm[19],
                    prm[20], prm[21], prm[22], prm[23] };

    float* wsf     = (float*)d_ws;
    float* d_dens  = wsf;                 // 16384
    float* d_ds    = wsf + 16384;         // 16384
    float* d_stats = wsf + 32768;         // 64 raw sums
    float* d_part  = wsf + 32832;         // 512 blocks * 32

    float* outp = (float*)d_out;

    density_kernel<<<BB * (NN / 64), 128, 0, stream>>>(xyz, d_dens);
    dn_mlp_kernel<<<1, 512, 0, stream>>>(d_dens, d_ds, dp);

    const int nblk = PKP / 1024;  // 512 blocks, 4 positions per thread
    wn_stats_kernel<0><<<nblk, 256, 0, stream>>>(gxn, idx, d_ds, d_stats, d_part, wp);
    wn_reduce_kernel<<<1, 64, 0, stream>>>(d_part, d_stats, 0, 16, nblk);
    wn_stats_kernel<1><<<nblk, 256, 0, stream>>>(gxn, idx, d_ds, d_stats, d_part, wp);
    wn_reduce_kernel<<<1, 64, 0, stream>>>(d_part, d_stats, 16, 16, nblk);
    wn_stats_kernel<2><<<nblk, 256, 0, stream>>>(gxn, idx, d_ds, d_stats, d_part, wp);
    wn_reduce_kernel<<<1, 64, 0, stream>>>(d_part, d_stats, 32, 32, nblk);

    final_einsum_kernel<<<BB * NN, 128, 0, stream>>>(gp, gxn, idx, d_ds, d_stats, outp, wp);
}